// MdTv2_26843545600413
// MI455X (gfx1250) — compile-verified
//
#include <hip/hip_runtime.h>
#include <math.h>

typedef __attribute__((ext_vector_type(2))) float v2f;
typedef __attribute__((ext_vector_type(8))) float v8f;

#define VOX 512000          // 80*80*80
#define CIN 32
#define DIM 48              // HEADS * HD
#define NHEADS 8
#define HD 6
#define LN_EPS 1e-5f

// ---------------------------------------------------------------------------
// Kernel 1: fused Linear(32->48) + LayerNorm via V_WMMA_F32_16X16X4_F32.
// grid.y = 0 -> F_feat -> qbuf ; grid.y = 1 -> M_feat -> kbuf.
// Each wave computes a 16-voxel x 48-channel tile: 3 N-tiles x 8 K-steps.
// ---------------------------------------------------------------------------
__global__ __launch_bounds__(256) void proj_ln_kernel(
    const float* __restrict__ F_feat, const float* __restrict__ M_feat,
    const float* __restrict__ proj_w, const float* __restrict__ proj_b,
    const float* __restrict__ ln_w,   const float* __restrict__ ln_b,
    float* __restrict__ qbuf, float* __restrict__ kbuf)
{
    const float* __restrict__ X = (blockIdx.y == 0) ? F_feat : M_feat;
    float* __restrict__ Y       = (blockIdx.y == 0) ? qbuf   : kbuf;

    __shared__ float tile[8][16 * DIM];   // 24 KB: one 16x48 tile per wave

    const int wave = threadIdx.x >> 5;
    const int lane = threadIdx.x & 31;
    const int lo   = lane & 15;           // voxel-in-tile (A rows) / N column (B)
    const int hi   = lane >> 4;           // K-pair select
    const int vbase = (blockIdx.x * 8 + wave) * 16;

    // Preload all of proj_w (48x32) in WMMA-B layout: B[k][n] = proj_w[n][k].
    // v0 holds K = 4s + 2*hi, v1 holds K = 4s + 2*hi + 1, column n = 16t + lo.
    v2f B[3][8];
#pragma unroll
    for (int t = 0; t < 3; ++t) {
#pragma unroll
        for (int s = 0; s < 8; ++s) {
            const int n = t * 16 + lo;
            const int k = s * 4 + 2 * hi;
            B[t][s].x = proj_w[n * CIN + k];
            B[t][s].y = proj_w[n * CIN + k + 1];
        }
    }

    v8f C0 = {}, C1 = {}, C2 = {};
#pragma unroll
    for (int s = 0; s < 8; ++s) {
        const int k = s * 4 + 2 * hi;
        v2f A;   // A[m][k]: lane lo = row m = voxel; channel-major input is coalesced
        A.x = X[(size_t)k       * VOX + vbase + lo];
        A.y = X[(size_t)(k + 1) * VOX + vbase + lo];
        C0 = __builtin_amdgcn_wmma_f32_16x16x4_f32(false, A, false, B[0][s],
                                                   (short)0, C0, false, false);
        C1 = __builtin_amdgcn_wmma_f32_16x16x4_f32(false, A, false, B[1][s],
                                                   (short)0, C1, false, false);
        C2 = __builtin_amdgcn_wmma_f32_16x16x4_f32(false, A, false, B[2][s],
                                                   (short)0, C2, false, false);
    }

    // Spill accumulators to LDS per documented C/D layout:
    // vgpr r, lanes 0-15 -> row r, lanes 16-31 -> row r+8; column = 16t + lo.
    float* tw = tile[wave];
#pragma unroll
    for (int r = 0; r < 8; ++r) {
        const int row = r + 8 * hi;
        tw[row * DIM +  0 + lo] = C0[r];
        tw[row * DIM + 16 + lo] = C1[r];
        tw[row * DIM + 32 + lo] = C2[r];
    }
    __syncthreads();

    // LayerNorm: lanes 0-15 each normalize one 48-channel row, write voxel-major.
    if (lo == lane) {   // lane < 16
        float y[DIM];
        float mu = 0.f;
#pragma unroll
        for (int c = 0; c < DIM; ++c) {
            y[c] = tw[lane * DIM + c] + proj_b[c];
            mu += y[c];
        }
        mu *= (1.f / DIM);
        float var = 0.f;
#pragma unroll
        for (int c = 0; c < DIM; ++c) {
            const float d = y[c] - mu;
            var += d * d;
        }
        const float rstd = rsqrtf(var * (1.f / DIM) + LN_EPS);
        float* __restrict__ yrow = Y + (size_t)(vbase + lane) * DIM;
#pragma unroll
        for (int c = 0; c < DIM; ++c)
            yrow[c] = (y[c] - mu) * rstd * ln_w[c] + ln_b[c];
    }
}

// ---------------------------------------------------------------------------
// Kernel 2: 27-neighborhood attention -> softmax -> motion + score.
// One thread per voxel. kbuf (98 MB) fits in the 192 MB L2, so the 27x reuse
// of neighbor reads stays on-chip.
// ---------------------------------------------------------------------------
__global__ __launch_bounds__(256) void attn_kernel(
    const float* __restrict__ qbuf, const float* __restrict__ kbuf,
    const float* __restrict__ rpb, float* __restrict__ out,
    float* __restrict__ score)
{
    const int v = blockIdx.x * 256 + threadIdx.x;     // grid is exact
    const int t = v % 80;
    const int w = (v / 80) % 80;
    const int h = v / 6400;

    const float* __restrict__ qv = qbuf + (size_t)v * DIM;
    float score_local = 0.f;

    for (int head = 0; head < NHEADS; ++head) {
        float qh[HD];
#pragma unroll
        for (int d = 0; d < HD; ++d) qh[d] = qv[head * HD + d];

        float attn[27];
        float maxv = -1e30f;
#pragma unroll
        for (int di = 0; di < 3; ++di) {
#pragma unroll
            for (int dj = 0; dj < 3; ++dj) {
#pragma unroll
                for (int dl = 0; dl < 3; ++dl) {
                    const int kk = di * 9 + dj * 3 + dl;
                    const int hh = h + di - 1, ww = w + dj - 1, tt = t + dl - 1;
                    float dot = 0.f;
                    if (((unsigned)hh < 80u) & ((unsigned)ww < 80u) & ((unsigned)tt < 80u)) {
                        const float* __restrict__ kv =
                            kbuf + (size_t)((hh * 80 + ww) * 80 + tt) * DIM + head * HD;
#pragma unroll
                        for (int d = 0; d < HD; ++d)
                            dot = fmaf(qh[d], kv[d], dot);
                    }
                    dot += rpb[head * 27 + kk];
                    attn[kk] = dot;
                    maxv = fmaxf(maxv, dot);
                }
            }
        }
        score_local += attn[14];   // K^3//2 + 1, pre-softmax

        float sum = 0.f;
#pragma unroll
        for (int kk = 0; kk < 27; ++kk) {
            const float e = __expf(attn[kk] - maxv);
            attn[kk] = e;
            sum += e;
        }
        const float inv = 1.f / sum;

        float m0 = 0.f, m1 = 0.f, m2 = 0.f;
#pragma unroll
        for (int di = 0; di < 3; ++di) {
#pragma unroll
            for (int dj = 0; dj < 3; ++dj) {
#pragma unroll
                for (int dl = 0; dl < 3; ++dl) {
                    const float a = attn[di * 9 + dj * 3 + dl];
                    m0 += a * (float)(di - 1);
                    m1 += a * (float)(dj - 1);
                    m2 += a * (float)(dl - 1);
                }
            }
        }
        out[((size_t)(head * 3 + 0)) * VOX + v] = m0 * inv;
        out[((size_t)(head * 3 + 1)) * VOX + v] = m1 * inv;
        out[((size_t)(head * 3 + 2)) * VOX + v] = m2 * inv;
    }

    // Block-level score reduction -> one atomic per block.
    __shared__ float red[256];
    red[threadIdx.x] = score_local;
    __syncthreads();
#pragma unroll
    for (int s = 128; s > 0; s >>= 1) {
        if (threadIdx.x < s) red[threadIdx.x] += red[threadIdx.x + s];
        __syncthreads();
    }
    if (threadIdx.x == 0)
        atomicAdd(score, red[0] * (1.0f / ((float)NHEADS * (float)VOX)));
}

__global__ void zero_score_kernel(float* __restrict__ score) { *score = 0.f; }

// ---------------------------------------------------------------------------
extern "C" void kernel_launch(void* const* d_in, const int* in_sizes, int n_in,
                              void* d_out, int out_size, void* d_ws, size_t ws_size,
                              hipStream_t stream)
{
    const float* F_feat = (const float*)d_in[0];
    const float* M_feat = (const float*)d_in[1];
    const float* proj_w = (const float*)d_in[2];
    const float* proj_b = (const float*)d_in[3];
    const float* ln_w   = (const float*)d_in[4];
    const float* ln_b   = (const float*)d_in[5];
    const float* rpb    = (const float*)d_in[6];

    float* out   = (float*)d_out;                    // 24 * VOX motion floats
    float* score = out + (size_t)3 * NHEADS * VOX;   // +1 scalar score
    float* qbuf  = (float*)d_ws;                     // VOX*48 f32
    float* kbuf  = qbuf + (size_t)VOX * DIM;         // VOX*48 f32

    zero_score_kernel<<<1, 1, 0, stream>>>(score);

    // 512000 voxels / (8 waves * 16 voxels) = 4000 blocks; y: {F->q, M->k}
    dim3 pgrid(4000, 2, 1);
    proj_ln_kernel<<<pgrid, 256, 0, stream>>>(F_feat, M_feat, proj_w, proj_b,
                                              ln_w, ln_b, qbuf, kbuf);

    attn_kernel<<<VOX / 256, 256, 0, stream>>>(qbuf, kbuf, rpb, out, score);
}